// NeighborSearch_39702677684834
// MI455X (gfx1250) — compile-verified
//
#include <hip/hip_runtime.h>

typedef __attribute__((ext_vector_type(2))) float v2f;
typedef __attribute__((ext_vector_type(8))) float v8f;
typedef int v4i __attribute__((vector_size(4 * sizeof(int))));

// Problem constants (reference: N_DATA = N_QUERY = 16384, DIM = 3)
#define NQ 16384
#define ND 16384
#define CHUNK 1024                 // data points staged per LDS buffer
#define NCHUNK (ND / CHUNK)        // 16
#define TILES_PER_CHUNK (CHUNK / 16)  // 64
#define WG_THREADS 256             // 8 wave32 -> 128 queries per workgroup
#define F4_PER_CHUNK (CHUNK * 3 / 4)  // 768 float4 per buffer (12KB)

// ---------------------------------------------------------------------------
// Optional CDNA5 async global->LDS path (GLOBAL_LOAD_ASYNC_TO_LDS_B128,
// tracked by ASYNCcnt). Falls back to a software-pipelined reg-staging copy.
// Builtin signature (from the previous round's diagnostic): param0 is a
// global-AS pointer to a 16-byte int vector; param1 the LDS-AS counterpart.
// ---------------------------------------------------------------------------
#if __has_builtin(__builtin_amdgcn_global_load_async_to_lds_b128)
#define HAVE_ASYNC_LDS 1
#else
#define HAVE_ASYNC_LDS 0
#endif

typedef __attribute__((address_space(1))) v4i* as1_v4i;
typedef __attribute__((address_space(3))) v4i* as3_v4i;

static __device__ __forceinline__ as1_v4i to_global_v4i(const void* p) {
    return (as1_v4i)(unsigned long long)(size_t)p;
}
static __device__ __forceinline__ as3_v4i to_lds_v4i(void* p) {
    // generic LDS address carries the LDS byte offset in its low 32 bits
    return (as3_v4i)(unsigned)(size_t)p;
}

struct StageRegs { float4 v[3]; };

// Issue the copy of one 12KB chunk (768 float4, 3 per thread).
static __device__ __forceinline__ void stage_issue(StageRegs& s,
                                                   const float4* __restrict__ g,
                                                   float4* l, int tid) {
#if HAVE_ASYNC_LDS
    (void)s;
#pragma unroll
    for (int r = 0; r < 3; ++r)
        __builtin_amdgcn_global_load_async_to_lds_b128(
            to_global_v4i(g + tid + 256 * r), to_lds_v4i(l + tid + 256 * r),
            /*offset=*/0, /*cpol=*/0);
#else
    (void)l;
#pragma unroll
    for (int r = 0; r < 3; ++r) s.v[r] = g[tid + 256 * r];
#endif
}

// Make the staged chunk resident in LDS (called after the compute phase that
// the copy was overlapped with; a __syncthreads() must follow before reads).
static __device__ __forceinline__ void stage_commit(StageRegs& s, float4* l, int tid) {
#if HAVE_ASYNC_LDS
    (void)s; (void)l; (void)tid;
#if __has_builtin(__builtin_amdgcn_s_wait_asynccnt)
    __builtin_amdgcn_s_wait_asynccnt(0);
#else
    __asm__ volatile("s_wait_asynccnt 0x0" ::: "memory");
#endif
#else
#pragma unroll
    for (int r = 0; r < 3; ++r) l[tid + 256 * r] = s.v[r];
#endif
}

// ---------------------------------------------------------------------------
// One 16(query) x 16(data) tile of squared distances via a single
// V_WMMA_F32_16X16X4_F32, data sourced from LDS:
//   A[m][:] = [qx, qy, qz, |q|^2]      (16x4)
//   B[:][n] = [-2dx, -2dy, -2dz, 1]    (4x16)
//   C[m][n] = |d_n|^2
//   D = |q|^2 - 2 q.d + |d|^2  ==  squared Euclidean distance
// ---------------------------------------------------------------------------
__device__ __forceinline__ v8f sqdist_tile_lds(v2f a, const float* __restrict__ buf,
                                               int ntile, int lane) {
    const int n = ntile * 16 + (lane & 15);
    const float dx = buf[n * 3 + 0];
    const float dy = buf[n * 3 + 1];
    const float dz = buf[n * 3 + 2];
    const float d2 = dx * dx + dy * dy + dz * dz;
    const bool lo = lane < 16;
    v2f b;
    b.x = lo ? (-2.0f * dx) : (-2.0f * dz);   // K0 row | K2 row
    b.y = lo ? (-2.0f * dy) : 1.0f;           // K1 row | K3 row
    v8f c = {d2, d2, d2, d2, d2, d2, d2, d2}; // |d|^2 depends only on N
    return __builtin_amdgcn_wmma_f32_16x16x4_f32(
        /*neg_a=*/false, a, /*neg_b=*/false, b,
        /*c_mod=*/(short)0, c, /*reuse_a=*/false, /*reuse_b=*/false);
}

__device__ __forceinline__ v2f make_a(const float* __restrict__ queries,
                                      int qbase, int lane) {
    const int m = qbase + (lane & 15);
    const float qx = queries[m * 3 + 0];
    const float qy = queries[m * 3 + 1];
    const float qz = queries[m * 3 + 2];
    const float q2 = qx * qx + qy * qy + qz * qz;
    const bool lo = lane < 16;
    v2f a;
    a.x = lo ? qx : qz;   // K0 | K2
    a.y = lo ? qy : q2;   // K1 | K3
    return a;
}

// ---------------------------------------------------------------------------
// Pass 1: per-query neighbor counts (int32) into d_ws.
// 8 waves/WG, one 16-row query tile per wave; data double-buffered in LDS.
// ---------------------------------------------------------------------------
__global__ void __launch_bounds__(WG_THREADS)
nsearch_count(const float* __restrict__ data,
              const float* __restrict__ queries,
              const float* __restrict__ radius_p,
              int* __restrict__ counts) {
    __shared__ float4 sbuf[2][F4_PER_CHUNK];
    const int tid = (int)threadIdx.x;
    const int lane = tid & 31;
    const int wave = tid >> 5;
    const int qbase = ((int)blockIdx.x * (WG_THREADS / 32) + wave) * 16;

    const float r = radius_p[0];
    const float r2 = r * r;
    const v2f a = make_a(queries, qbase, lane);
    const float4* gdata = (const float4*)data;

    const int myi = lane & 7;
    const int myhalf = (lane >> 3) & 1;
    int cnt = 0;

    StageRegs st;
    stage_issue(st, gdata, &sbuf[0][0], tid);
    stage_commit(st, &sbuf[0][0], tid);
    __syncthreads();

    for (int c = 0; c < NCHUNK; ++c) {
        const int pp = c & 1;
        const bool more = (c + 1) < NCHUNK;
        if (more) stage_issue(st, gdata + (c + 1) * F4_PER_CHUNK, &sbuf[pp ^ 1][0], tid);

        const float* buf = (const float*)&sbuf[pp][0];
#pragma unroll 2
        for (int t = 0; t < TILES_PER_CHUNK; ++t) {
            v8f d = sqdist_tile_lds(a, buf, t, lane);
#pragma unroll
            for (int i = 0; i < 8; ++i) {
                unsigned m = __builtin_amdgcn_ballot_w32(d[i] <= r2);
                if (myi == i) cnt += __popc((int)((m >> (myhalf * 16)) & 0xFFFFu));
            }
        }

        if (more) {
            stage_commit(st, &sbuf[pp ^ 1][0], tid);
            __syncthreads();
        }
    }
    if (lane < 16) counts[qbase + lane] = cnt;
}

// ---------------------------------------------------------------------------
// Pass 2: exclusive scan of 16384 counts -> int64 splits[16385].
// ---------------------------------------------------------------------------
__global__ void nsearch_scan(const int* __restrict__ counts,
                             long long* __restrict__ splits) {
    __shared__ int sums[1024];
    const int t = (int)threadIdx.x;
    const int base = t * 16;

    int local[16];
    int s = 0;
#pragma unroll
    for (int j = 0; j < 16; ++j) {
        local[j] = counts[base + j];
        s += local[j];
    }
    sums[t] = s;
    __syncthreads();

    for (int off = 1; off < 1024; off <<= 1) {
        int v = (t >= off) ? sums[t - off] : 0;
        __syncthreads();
        sums[t] += v;
        __syncthreads();
    }

    long long run = (t == 0) ? 0LL : (long long)sums[t - 1];
    if (t == 0) splits[0] = 0LL;
#pragma unroll
    for (int j = 0; j < 16; ++j) {
        run += (long long)local[j];
        splits[base + j + 1] = run;
    }
}

// ---------------------------------------------------------------------------
// Pass 3: write neighbor indices. Identical WMMA/LDS path as pass 1 so the
// mask is bit-exactly the same; lane L (L<16) drains its row's ballot half
// in ascending-bit (== ascending data index) order at cursor splits[q].
// ---------------------------------------------------------------------------
__global__ void __launch_bounds__(WG_THREADS)
nsearch_fill(const float* __restrict__ data,
             const float* __restrict__ queries,
             const float* __restrict__ radius_p,
             const long long* __restrict__ splits,
             long long* __restrict__ out_idx) {
    __shared__ float4 sbuf[2][F4_PER_CHUNK];
    const int tid = (int)threadIdx.x;
    const int lane = tid & 31;
    const int wave = tid >> 5;
    const int qbase = ((int)blockIdx.x * (WG_THREADS / 32) + wave) * 16;

    const float r = radius_p[0];
    const float r2 = r * r;
    const v2f a = make_a(queries, qbase, lane);
    const float4* gdata = (const float4*)data;

    const int myi = lane & 7;
    const int sh = ((lane >> 3) & 1) * 16;
    int cursor = 0;
    if (lane < 16) cursor = (int)splits[qbase + lane];

    StageRegs st;
    stage_issue(st, gdata, &sbuf[0][0], tid);
    stage_commit(st, &sbuf[0][0], tid);
    __syncthreads();

    for (int c = 0; c < NCHUNK; ++c) {
        const int pp = c & 1;
        const bool more = (c + 1) < NCHUNK;
        if (more) stage_issue(st, gdata + (c + 1) * F4_PER_CHUNK, &sbuf[pp ^ 1][0], tid);

        const float* buf = (const float*)&sbuf[pp][0];
        const int nchunkbase = c * CHUNK;
#pragma unroll 2
        for (int t = 0; t < TILES_PER_CHUNK; ++t) {
            v8f d = sqdist_tile_lds(a, buf, t, lane);
            unsigned mymask = 0;
#pragma unroll
            for (int i = 0; i < 8; ++i) {
                unsigned m = __builtin_amdgcn_ballot_w32(d[i] <= r2);
                if (myi == i) mymask = m;
            }
            if (lane < 16) {
                unsigned h = (mymask >> sh) & 0xFFFFu;
                const int nb = nchunkbase + t * 16;
                while (h) {
                    int b = (int)__builtin_ctz(h);
                    out_idx[cursor++] = (long long)(nb + b);
                    h &= h - 1u;
                }
            }
        }

        if (more) {
            stage_commit(st, &sbuf[pp ^ 1][0], tid);
            __syncthreads();
        }
    }
}

// ---------------------------------------------------------------------------
extern "C" void kernel_launch(void* const* d_in, const int* in_sizes, int n_in,
                              void* d_out, int out_size, void* d_ws, size_t ws_size,
                              hipStream_t stream) {
    (void)in_sizes; (void)n_in; (void)ws_size;
    const float* data    = (const float*)d_in[0];
    const float* queries = (const float*)d_in[1];
    const float* radius  = (const float*)d_in[2];

    long long* out = (long long*)d_out;
    // d_out = [ neighbors_index (K) | splits (NQ+1) ], K inferred from out_size.
    const int K = out_size - (NQ + 1);
    long long* splits = out + K;

    int* counts = (int*)d_ws; // NQ * 4 bytes of scratch

    // 128 queries per workgroup (8 wave32 x 16) -> 128 workgroups.
    const int blocks = NQ / ((WG_THREADS / 32) * 16);

    nsearch_count<<<blocks, WG_THREADS, 0, stream>>>(data, queries, radius, counts);
    nsearch_scan<<<1, 1024, 0, stream>>>(counts, splits);
    nsearch_fill<<<blocks, WG_THREADS, 0, stream>>>(data, queries, radius, splits, out);
}